// Region_86414741995824
// MI455X (gfx1250) — compile-verified
//
#include <hip/hip_runtime.h>
#include <stdint.h>

// Problem constants (match reference)
#define Bn     32
#define Rn     13
#define Cn     13
#define An     5
#define Kn     20
#define CELLn  25          // K + 5
#define Nn     845         // R*C*A
#define OUTC   25          // output channels per box
#define THRESH  0.5f
#define NMS_TH  0.4f

#define NMS_THREADS 256

// ---------------------------------------------------------------------------
// CDNA5 async global->LDS copy helpers (gfx1250 only; tracked by ASYNCcnt).
// LDS byte offset = low 32 bits of the generic pointer to __shared__ data
// (LDS aperture: addr[31:0] is the workgroup-relative LDS offset).
// ---------------------------------------------------------------------------
__device__ __forceinline__ void async_ld_f32(float* lds_dst, const float* gsrc) {
#if defined(__gfx1250__)
    unsigned int l = (unsigned int)(uintptr_t)lds_dst;
    asm volatile("global_load_async_to_lds_b32 %0, %1, off"
                 :: "v"(l), "v"(gsrc) : "memory");
#else
    *lds_dst = *gsrc;
#endif
}

__device__ __forceinline__ void async_wait0() {
#if defined(__gfx1250__)
    asm volatile("s_wait_asynccnt 0" ::: "memory");
#endif
}

__device__ __forceinline__ float sigmoidf_(float v) {
    return 1.0f / (1.0f + expf(-v));
}

// ---------------------------------------------------------------------------
// Kernel 1: decode.  One thread per (b, n) box.
//   out[b,n,0..4]   = bx, by, bw, bh, obj
//   out[b,n,5+k]    = thresholded prob (pre-NMS)
//   ws SoA (5 arrays of B*N floats): x1, y1, x2, y2, area
// ---------------------------------------------------------------------------
__global__ void region_decode_kernel(const float* __restrict__ x,
                                     const float* __restrict__ bias,
                                     float* __restrict__ out,
                                     float* __restrict__ boxes) {
    int idx = blockIdx.x * blockDim.x + threadIdx.x;
    if (idx >= Bn * Nn) return;
    int b = idx / Nn;
    int n = idx - b * Nn;
    int a  = n % An;
    int rc = n / An;        // r*C + c
    int c  = rc % Cn;
    int r  = rc / Cn;

    const float* t = x + ((size_t)b * (Rn * Cn) + rc) * (An * CELLn) + a * CELLn;

    float t0 = t[0], t1 = t[1], t2 = t[2], t3 = t[3], t4 = t[4];

    float bx = (sigmoidf_(t0) + (float)c) * (1.0f / (float)Cn);
    float by = (sigmoidf_(t1) + (float)r) * (1.0f / (float)Rn);
    float bw = expf(t2) * bias[2 * a]     / (float)Rn;   // matches reference (/R)
    float bh = expf(t3) * bias[2 * a + 1] / (float)Cn;   // matches reference (/C)
    float obj = sigmoidf_(t4);

    // softmax over classes (max-subtracted, like jax.nn.softmax)
    float cls[Kn];
    float m = t[5];
    #pragma unroll
    for (int k = 0; k < Kn; ++k) { cls[k] = t[5 + k]; m = fmaxf(m, cls[k]); }
    float s = 0.0f;
    #pragma unroll
    for (int k = 0; k < Kn; ++k) { cls[k] = expf(cls[k] - m); s += cls[k]; }
    float inv = 1.0f / s;

    float* orow = out + (size_t)idx * OUTC;
    orow[0] = bx; orow[1] = by; orow[2] = bw; orow[3] = bh; orow[4] = obj;
    #pragma unroll
    for (int k = 0; k < Kn; ++k) {
        float cn = cls[k] * inv;       // softmax value first (reference grouping)
        float p  = obj * cn;
        orow[5 + k] = (p > THRESH) ? p : 0.0f;
    }

    float x1 = bx - 0.5f * bw;
    float y1 = by - 0.5f * bh;
    const int BN = Bn * Nn;
    boxes[0 * BN + idx] = x1;
    boxes[1 * BN + idx] = y1;
    boxes[2 * BN + idx] = x1 + bw;
    boxes[3 * BN + idx] = y1 + bh;
    boxes[4 * BN + idx] = bw * bh;
}

// ---------------------------------------------------------------------------
// Kernel 2: greedy per-class NMS. One block per (b,k); 640 blocks x 256 thr.
// Box table + score column staged into LDS via CDNA5 async global->LDS copies.
// Exactly reproduces reference semantics: process in descending-score order
// (ties -> lower index, matching stable argsort of -scores); a chosen box
// suppresses all still-alive boxes with IoU > 0.4; suppressed boxes never
// suppress others.  state: 0=invalid, 1=alive, 2=kept, 3=suppressed.
// ---------------------------------------------------------------------------
__global__ void region_nms_kernel(const float* __restrict__ boxes,
                                  float* __restrict__ out) {
    __shared__ float sx1[Nn], sy1[Nn], sx2[Nn], sy2[Nn], sar[Nn], ssc[Nn];
    __shared__ unsigned char st[Nn];
    __shared__ float red_s[NMS_THREADS];
    __shared__ int   red_i[NMS_THREADS];

    const int b   = blockIdx.x / Kn;
    const int k   = blockIdx.x - b * Kn;
    const int tid = threadIdx.x;
    const int BN  = Bn * Nn;

    const float* gx1 = boxes + 0 * BN + b * Nn;
    const float* gy1 = boxes + 1 * BN + b * Nn;
    const float* gx2 = boxes + 2 * BN + b * Nn;
    const float* gy2 = boxes + 3 * BN + b * Nn;
    const float* gar = boxes + 4 * BN + b * Nn;

    // Stage box table + this class's score column into LDS (async DMA path).
    for (int n = tid; n < Nn; n += NMS_THREADS) {
        async_ld_f32(&sx1[n], &gx1[n]);
        async_ld_f32(&sy1[n], &gy1[n]);
        async_ld_f32(&sx2[n], &gx2[n]);
        async_ld_f32(&sy2[n], &gy2[n]);
        async_ld_f32(&sar[n], &gar[n]);
        async_ld_f32(&ssc[n], out + ((size_t)b * Nn + n) * OUTC + 5 + k);
    }
    async_wait0();          // my wave's async copies have landed in LDS
    __syncthreads();        // everyone's copies have landed

    for (int n = tid; n < Nn; n += NMS_THREADS)
        st[n] = (ssc[n] > THRESH) ? (unsigned char)1 : (unsigned char)0;
    __syncthreads();

    // Greedy loop: each iteration keeps the highest-scoring alive box.
    for (;;) {
        // per-thread argmax over owned (strided) indices; ascending scan with
        // strict '>' keeps the lowest index among per-thread ties
        float bs = -1.0f; int bi = -1;
        for (int n = tid; n < Nn; n += NMS_THREADS) {
            if (st[n] == 1) {
                float sc = ssc[n];
                if (sc > bs) { bs = sc; bi = n; }
            }
        }
        red_s[tid] = bs; red_i[tid] = bi;
        __syncthreads();
        // tree reduction, tie -> lower index
        for (int off = NMS_THREADS / 2; off > 0; off >>= 1) {
            if (tid < off) {
                float os = red_s[tid + off]; int oi = red_i[tid + off];
                float cs = red_s[tid];       int ci = red_i[tid];
                if ((os > cs) || (os == cs && oi >= 0 && (ci < 0 || oi < ci))) {
                    red_s[tid] = os; red_i[tid] = oi;
                }
            }
            __syncthreads();
        }
        int best = red_i[0];
        if (best < 0) break;            // uniform across block

        if (tid == 0) st[best] = 2;     // keep it (no other thread touches st[best])

        float bx1 = sx1[best], by1 = sy1[best];
        float bx2 = sx2[best], by2 = sy2[best];
        float bar = sar[best];

        for (int n = tid; n < Nn; n += NMS_THREADS) {
            if (n == best || st[n] != 1) continue;
            float iw = fmaxf(0.0f, fminf(sx2[n], bx2) - fmaxf(sx1[n], bx1));
            float ih = fmaxf(0.0f, fminf(sy2[n], by2) - fmaxf(sy1[n], by1));
            float inter = iw * ih;
            float uni   = sar[n] + bar - inter;
            float iou   = inter / fmaxf(uni, 1e-9f);
            if (iou > NMS_TH) st[n] = 3;
        }
        __syncthreads();
    }

    // keep = kept; everything else (invalid/suppressed) -> 0
    for (int n = tid; n < Nn; n += NMS_THREADS) {
        float v = (st[n] == 2) ? ssc[n] : 0.0f;
        out[((size_t)b * Nn + n) * OUTC + 5 + k] = v;
    }
}

// ---------------------------------------------------------------------------
extern "C" void kernel_launch(void* const* d_in, const int* in_sizes, int n_in,
                              void* d_out, int out_size, void* d_ws, size_t ws_size,
                              hipStream_t stream) {
    (void)in_sizes; (void)n_in; (void)out_size; (void)ws_size;
    const float* x    = (const float*)d_in[0];
    const float* bias = (const float*)d_in[1];
    float* out   = (float*)d_out;
    float* boxes = (float*)d_ws;    // needs 5 * B * N * 4 = 540,800 bytes

    const int total  = Bn * Nn;                     // 27040 boxes
    const int dblk   = 256;
    const int dgrid  = (total + dblk - 1) / dblk;   // 106 blocks

    region_decode_kernel<<<dgrid, dblk, 0, stream>>>(x, bias, out, boxes);
    region_nms_kernel<<<Bn * Kn, NMS_THREADS, 0, stream>>>(boxes, out);
}